// MoE_58892591563429
// MI455X (gfx1250) — compile-verified
//
#include <hip/hip_runtime.h>
#include <hip/hip_bf16.h>
#include <math.h>

#define B_TOK 16384
#define D_IN  128
#define H_DIM 512
#define O_DIM 128
#define E_NUM 8
#define TM    32            // tokens per expert tile (2 x 16-row WMMA tiles)

typedef __attribute__((ext_vector_type(2))) float v2f;
typedef __attribute__((ext_vector_type(8))) float v8f;

// ---------------------------------------------------------------------------
// 0) init: zero output accumulator + per-expert counters/importance
// ---------------------------------------------------------------------------
__global__ __launch_bounds__(256) void moe_init(float* __restrict__ y,
                                                int* __restrict__ cnt,
                                                float* __restrict__ importance) {
  int idx = blockIdx.x * 256 + threadIdx.x;           // float4 index, exactly covers B*O
  ((float4*)y)[idx] = make_float4(0.f, 0.f, 0.f, 0.f);
  if (blockIdx.x == 0) {
    if (threadIdx.x < E_NUM) cnt[threadIdx.x] = 0;
    else if (threadIdx.x < 2 * E_NUM) importance[threadIdx.x - E_NUM] = 0.f;
  }
}

// ---------------------------------------------------------------------------
// 1) gating: logits = x @ w_gate ; top-2 softmax ; bucket tokens per expert
// ---------------------------------------------------------------------------
__global__ __launch_bounds__(256) void moe_gate(const float* __restrict__ x,
                                                const float* __restrict__ w_gate,
                                                float* __restrict__ gates,
                                                int* __restrict__ bucket,
                                                int* __restrict__ cnt,
                                                float* __restrict__ importance) {
  __shared__ float wg[D_IN * E_NUM];   // 4 KB
  __shared__ float simp[E_NUM];
  const int tid = threadIdx.x;
  for (int i = tid; i < D_IN * E_NUM; i += 256) wg[i] = w_gate[i];
  if (tid < E_NUM) simp[tid] = 0.f;
  __syncthreads();

  const int b = blockIdx.x * 256 + tid;
  const float4* xr = (const float4*)(x + (size_t)b * D_IN);
  float l[E_NUM];
#pragma unroll
  for (int e = 0; e < E_NUM; ++e) l[e] = 0.f;
#pragma unroll 4
  for (int d4 = 0; d4 < D_IN / 4; ++d4) {
    float4 xv = xr[d4];
    const float* wr = &wg[d4 * 4 * E_NUM];
#pragma unroll
    for (int e = 0; e < E_NUM; ++e)
      l[e] += xv.x * wr[e] + xv.y * wr[E_NUM + e] + xv.z * wr[2 * E_NUM + e] + xv.w * wr[3 * E_NUM + e];
  }

  // top-2 (stable: lowest index wins ties, matching lax.top_k)
  int e0 = 0; float l0 = l[0];
#pragma unroll
  for (int e = 1; e < E_NUM; ++e) if (l[e] > l0) { l0 = l[e]; e0 = e; }
  int e1 = -1; float l1 = -3.4e38f;
#pragma unroll
  for (int e = 0; e < E_NUM; ++e) if (e != e0 && l[e] > l1) { l1 = l[e]; e1 = e; }

  float q  = __expf(l1 - l0);       // stable 2-way softmax, T=1
  float g0 = 1.f / (1.f + q);
  float g1 = q / (1.f + q);

  float gr[E_NUM];
#pragma unroll
  for (int e = 0; e < E_NUM; ++e) gr[e] = 0.f;
  gr[e0] = g0; gr[e1] = g1;
#pragma unroll
  for (int e = 0; e < E_NUM; ++e) gates[(size_t)b * E_NUM + e] = gr[e];

  atomicAdd(&simp[e0], g0);         // ds_add_f32, block-local reduction
  atomicAdd(&simp[e1], g1);
  int p0 = atomicAdd(&cnt[e0], 1); bucket[e0 * B_TOK + p0] = b;
  int p1 = atomicAdd(&cnt[e1], 1); bucket[e1 * B_TOK + p1] = b;
  __syncthreads();
  if (tid < E_NUM) unsafeAtomicAdd(&importance[tid], simp[tid]);
}

// ---------------------------------------------------------------------------
// 2) loss: cv_squared(importance) + cv_squared(load), ddof=1
// ---------------------------------------------------------------------------
__global__ void moe_loss(const float* __restrict__ importance,
                         const int* __restrict__ cnt,
                         float* __restrict__ out_loss) {
  if (threadIdx.x == 0 && blockIdx.x == 0) {
    float im[E_NUM], ld[E_NUM], mi = 0.f, ml = 0.f;
    for (int e = 0; e < E_NUM; ++e) {
      im[e] = importance[e]; ld[e] = (float)cnt[e];
      mi += im[e]; ml += ld[e];
    }
    mi *= (1.f / E_NUM); ml *= (1.f / E_NUM);
    float vi = 0.f, vl = 0.f;
    for (int e = 0; e < E_NUM; ++e) {
      float a = im[e] - mi, c = ld[e] - ml;
      vi += a * a; vl += c * c;
    }
    vi *= (1.f / (E_NUM - 1)); vl *= (1.f / (E_NUM - 1));
    *out_loss = vi / (mi * mi + 1e-10f) + vl / (ml * ml + 1e-10f);
  }
}

// ---------------------------------------------------------------------------
// 3) experts: per (expert, 32-token tile): relu(x@w1+b1)@w2+b2, gated scatter.
//    Each B fragment (w1/w2 from L2) feeds TWO WMMAs (M-tiles 0..15 / 16..31).
// ---------------------------------------------------------------------------
__global__ __launch_bounds__(256) void moe_expert(const float* __restrict__ x,
                                                  const float* __restrict__ w1,
                                                  const float* __restrict__ b1,
                                                  const float* __restrict__ w2,
                                                  const float* __restrict__ b2,
                                                  const float* __restrict__ gates,
                                                  const int* __restrict__ bucket,
                                                  const int* __restrict__ cnt,
                                                  float* __restrict__ y) {
  __shared__ float xs[TM][132];   // +4 pad -> conflict-free column b64 reads (16.9 KB)
  __shared__ float hs[TM][516];   // +4 pad (66 KB)  -- total ~83 KB of 320 KB WGP LDS
  __shared__ float gbuf[TM];
  __shared__ int   tokbuf[TM];

  const int e    = blockIdx.y;
  const int n    = cnt[e];
  const int base = blockIdx.x * TM;
  if (base >= n) return;                 // uniform early exit
  const int m = min(TM, n - base);

  const int tid  = threadIdx.x;
  const int lane = tid & 31;
  const int w    = tid >> 5;             // wave id 0..7
  const int lh   = lane >> 4;            // half-wave (K-pair select)
  const int ln   = lane & 15;            // row / column within 16

  if (tid < TM) {
    int idx = base + tid;
    int tok = bucket[e * B_TOK + ((idx < n) ? idx : (n - 1))];
    tokbuf[tid] = tok;
    gbuf[tid]   = (idx < n) ? gates[(size_t)tok * E_NUM + e] : 0.f;
  }
  __syncthreads();

  // gather 32 token rows of x into LDS (coalesced 128-float runs)
#pragma unroll
  for (int i = 0; i < TM * D_IN / 256; ++i) {
    int idx = tid + i * 256;             // 0..4095
    int r = idx >> 7, c = idx & 127;
    xs[r][c] = x[(size_t)tokbuf[r] * D_IN + c];
  }
  __syncthreads();

  // ---- GEMM1: h[32x512] = relu(x_tile @ w1[e] + b1[e]); wave w -> 64 H cols
  {
    const float* w1e = w1 + (size_t)e * D_IN * H_DIM;
    const float* pB  = w1e + (2 * lh) * H_DIM + w * 64 + ln;
    v8f acc[2][4];                       // [mtile][ntile]
#pragma unroll
    for (int nt = 0; nt < 4; ++nt) {
      float bv = b1[e * H_DIM + w * 64 + nt * 16 + ln];
#pragma unroll
      for (int i = 0; i < 8; ++i) { acc[0][nt][i] = bv; acc[1][nt][i] = bv; }
    }
    for (int kk = 0; kk < D_IN; kk += 4) {
      float2 a2l = *(const float2*)&xs[ln][kk + 2 * lh];        // rows 0..15
      float2 a2h = *(const float2*)&xs[16 + ln][kk + 2 * lh];   // rows 16..31
      v2f a0; a0[0] = a2l.x; a0[1] = a2l.y;
      v2f a1; a1[0] = a2h.x; a1[1] = a2h.y;
#pragma unroll
      for (int nt = 0; nt < 4; ++nt) {
        v2f bb;
        bb[0] = pB[kk * H_DIM + nt * 16];
        bb[1] = pB[kk * H_DIM + nt * 16 + H_DIM];
        acc[0][nt] = __builtin_amdgcn_wmma_f32_16x16x4_f32(false, a0, false, bb,
                                                           (short)0, acc[0][nt], false, false);
        acc[1][nt] = __builtin_amdgcn_wmma_f32_16x16x4_f32(false, a1, false, bb,
                                                           (short)0, acc[1][nt], false, false);
      }
    }
#pragma unroll
    for (int mt = 0; mt < 2; ++mt)
#pragma unroll
      for (int nt = 0; nt < 4; ++nt)
#pragma unroll
        for (int i = 0; i < 8; ++i)
          hs[mt * 16 + 8 * lh + i][w * 64 + nt * 16 + ln] = fmaxf(acc[mt][nt][i], 0.f);
  }
  __syncthreads();

  // ---- GEMM2: y[32x128] = h @ w2[e] + b2[e]; wave w -> 16 O cols, K=512.
  //      4 independent accumulator chains: [mtile][k-parity].
  {
    const int o0 = w * 16;
    const float* w2e = w2 + (size_t)e * H_DIM * O_DIM;
    const float* pB  = w2e + (2 * lh) * O_DIM + o0 + ln;
    v8f acc[2][2];
    float bv = b2[e * O_DIM + o0 + ln];
#pragma unroll
    for (int i = 0; i < 8; ++i) {
      acc[0][0][i] = bv;  acc[0][1][i] = 0.f;
      acc[1][0][i] = bv;  acc[1][1][i] = 0.f;
    }
    for (int kk = 0; kk < H_DIM; kk += 8) {
#pragma unroll
      for (int p = 0; p < 2; ++p) {
        int koff = kk + 4 * p;
        float2 a2l = *(const float2*)&hs[ln][koff + 2 * lh];
        float2 a2h = *(const float2*)&hs[16 + ln][koff + 2 * lh];
        v2f a0; a0[0] = a2l.x; a0[1] = a2l.y;
        v2f a1; a1[0] = a2h.x; a1[1] = a2h.y;
        v2f bb;
        bb[0] = pB[koff * O_DIM];
        bb[1] = pB[koff * O_DIM + O_DIM];
        acc[0][p] = __builtin_amdgcn_wmma_f32_16x16x4_f32(false, a0, false, bb,
                                                          (short)0, acc[0][p], false, false);
        acc[1][p] = __builtin_amdgcn_wmma_f32_16x16x4_f32(false, a1, false, bb,
                                                          (short)0, acc[1][p], false, false);
      }
    }
#pragma unroll
    for (int mt = 0; mt < 2; ++mt)
#pragma unroll
      for (int i = 0; i < 8; ++i) {
        int M = mt * 16 + 8 * lh + i;
        float v = (acc[mt][0][i] + acc[mt][1][i]) * gbuf[M];
        if (M < m)
          unsafeAtomicAdd(&y[(size_t)tokbuf[M] * O_DIM + o0 + ln], v);
      }
  }
}

// ---------------------------------------------------------------------------
// 4) combine() zero-fill: y==0 -> eps
// ---------------------------------------------------------------------------
__global__ __launch_bounds__(256) void moe_eps(float* __restrict__ y) {
  const float EPSV = 2.2204460492503131e-16f;
  int idx = blockIdx.x * 256 + threadIdx.x;   // float4 index
  float4* p = ((float4*)y) + idx;
  float4 v = *p;
  v.x = (v.x == 0.f) ? EPSV : v.x;
  v.y = (v.y == 0.f) ? EPSV : v.y;
  v.z = (v.z == 0.f) ? EPSV : v.z;
  v.w = (v.w == 0.f) ? EPSV : v.w;
  *p = v;
}

// ---------------------------------------------------------------------------
extern "C" void kernel_launch(void* const* d_in, const int* in_sizes, int n_in,
                              void* d_out, int out_size, void* d_ws, size_t ws_size,
                              hipStream_t stream) {
  (void)in_sizes; (void)n_in; (void)out_size; (void)ws_size;
  const float* x      = (const float*)d_in[0];
  const float* w_gate = (const float*)d_in[1];
  const float* w1     = (const float*)d_in[2];
  const float* b1     = (const float*)d_in[3];
  const float* w2     = (const float*)d_in[4];
  const float* b2     = (const float*)d_in[5];

  float* y    = (float*)d_out;
  float* loss = y + (size_t)B_TOK * O_DIM;

  char*  ws         = (char*)d_ws;
  float* gates      = (float*)ws;                                   // B*E f32   (512 KB)
  int*   bucket     = (int*)(ws + (size_t)B_TOK * E_NUM * 4);       // E*B i32   (512 KB)
  int*   cnt        = (int*)(ws + 2 * (size_t)B_TOK * E_NUM * 4);   // E i32
  float* importance = (float*)(ws + 2 * (size_t)B_TOK * E_NUM * 4 + 64);

  moe_init<<<(B_TOK * O_DIM) / (4 * 256), 256, 0, stream>>>(y, cnt, importance);
  moe_gate<<<B_TOK / 256, 256, 0, stream>>>(x, w_gate, gates, bucket, cnt, importance);
  moe_loss<<<1, 32, 0, stream>>>(importance, cnt, loss);
  dim3 g(B_TOK / TM, E_NUM);
  moe_expert<<<g, 256, 0, stream>>>(x, w1, b1, w2, b2, gates, bucket, cnt, y);
  moe_eps<<<(B_TOK * O_DIM) / (4 * 256), 256, 0, stream>>>(y);
}